// SecondaryCaps_38835094290470
// MI455X (gfx1250) — compile-verified
//
#include <hip/hip_runtime.h>
#include <hip/hip_bf16.h>
#include <math.h>

typedef __attribute__((ext_vector_type(2))) float v2f;
typedef __attribute__((ext_vector_type(8))) float v8f;

#define BATCH 64
#define IN_C  2048
#define OUT_C 32
#define OUT_D 16
#define IN_D  8
#define NI    16   // input capsules per block chunk

static __device__ __forceinline__ v8f wmma_f32_16x16x4(v2f a, v2f b, v8f c) {
  // (neg_a, A, neg_b, B, c_mod, C, reuse_a, reuse_b)
  return __builtin_amdgcn_wmma_f32_16x16x4_f32(false, a, false, b, (short)0, c,
                                               false, false);
}

// ---- softmax norms over i for each (b,o): max and 1/sum(exp) --------------
__global__ void caps_norms(const float* __restrict__ blog,
                           float* __restrict__ mx_out,
                           float* __restrict__ invz_out) {
  const int bo = blockIdx.x;            // 0..2047 = b*32+o
  const int b  = bo >> 5;
  const int o  = bo & 31;
  const int tid = threadIdx.x;          // 128 threads
  __shared__ float red[128];
  const float* p = blog + ((size_t)b * IN_C) * OUT_C + o;
  float loc[IN_C / 128];
#pragma unroll
  for (int k = 0; k < IN_C / 128; ++k)
    loc[k] = p[(size_t)(tid + 128 * k) * OUT_C];
  float m = loc[0];
#pragma unroll
  for (int k = 1; k < IN_C / 128; ++k) m = fmaxf(m, loc[k]);
  red[tid] = m;
  __syncthreads();
  for (int st = 64; st > 0; st >>= 1) {
    if (tid < st) red[tid] = fmaxf(red[tid], red[tid + st]);
    __syncthreads();
  }
  m = red[0];
  __syncthreads();
  float sum = 0.f;
#pragma unroll
  for (int k = 0; k < IN_C / 128; ++k) sum += expf(loc[k] - m);
  red[tid] = sum;
  __syncthreads();
  for (int st = 64; st > 0; st >>= 1) {
    if (tid < st) red[tid] += red[tid + st];
    __syncthreads();
  }
  if (tid == 0) {
    mx_out[bo]   = m;
    invz_out[bo] = 1.0f / red[0];
  }
}

// ---- s[b,o,d] += sum_i c[b,i,o] * u_hat[b,i,o,d], u_hat via WMMA ----------
__global__ void __launch_bounds__(256)
caps_s_wmma(const float* __restrict__ x, const float* __restrict__ Wm,
            const float* __restrict__ blog, const float* __restrict__ mx,
            const float* __restrict__ invz, float* __restrict__ s_out) {
  const int tid  = threadIdx.x;
  const int wave = tid >> 5;
  const int lane = tid & 31;
  const int h    = lane >> 4;   // wave half
  const int l15  = lane & 15;
  const int b0   = blockIdx.x * 16;
  const int i0   = blockIdx.y * NI;

  __shared__ float cw[NI * 16 * 33];  // [ii][b_local][o], stride 33 (bank pad)
  for (int n = tid; n < NI * 16 * 32; n += 256) {
    const int ii = n >> 9;
    const int bl = (n >> 5) & 15;
    const int o  = n & 31;
    const int b  = b0 + bl;
    const float bv = blog[((size_t)b * IN_C + (i0 + ii)) * OUT_C + o];
    const int bo = b * OUT_C + o;
    cw[(ii * 16 + bl) * 33 + o] = expf(bv - mx[bo]) * invz[bo];
  }
  __syncthreads();

  v8f acc[4] = {{}, {}, {}, {}};      // one o per tile (od-tile == capsule o)
  const int o_base = wave * 4;        // 8 waves x 4 = 32 capsules
  for (int ii = 0; ii < NI; ++ii) {
    const int i = i0 + ii;
    const float* xp = x + ((size_t)(b0 + l15) * IN_C + i) * IN_D;
    const v2f a_lo = *(const v2f*)(xp + 2 * h);       // K = 2h, 2h+1
    const v2f a_hi = *(const v2f*)(xp + 4 + 2 * h);   // K = 4+2h, 5+2h
    const float* cwrow = &cw[(ii * 16 + l15) * 33];
#pragma unroll
    for (int j = 0; j < 4; ++j) {
      const int o = o_base + j;
      const float c = cwrow[o];       // fold routing coeff into A rows
      const v2f al = a_lo * c;
      const v2f ah = a_hi * c;
      const float* wp = Wm + (((size_t)i * OUT_C + o) * OUT_D + l15) * IN_D;
      const v2f b_lo = *(const v2f*)(wp + 2 * h);
      const v2f b_hi = *(const v2f*)(wp + 4 + 2 * h);
      acc[j] = wmma_f32_16x16x4(al, b_lo, acc[j]);
      acc[j] = wmma_f32_16x16x4(ah, b_hi, acc[j]);
    }
  }
#pragma unroll
  for (int j = 0; j < 4; ++j) {
    const int o = o_base + j;
#pragma unroll
    for (int v = 0; v < 8; ++v) {     // C row M = v + 8h
      const int b = b0 + v + 8 * h;
      atomicAdd(&s_out[((size_t)b * OUT_C + o) * OUT_D + l15], acc[j][v]);
    }
  }
}

// ---- blog[b,i,o] += sum_d u_hat[b,i,o,d] * v[b,o,d] -----------------------
__global__ void __launch_bounds__(256)
caps_b_wmma(const float* __restrict__ x, const float* __restrict__ Wm,
            const float* __restrict__ vprev, float* __restrict__ blog) {
  const int tid  = threadIdx.x;
  const int wave = tid >> 5;
  const int lane = tid & 31;
  const int h    = lane >> 4;
  const int l15  = lane & 15;
  const int b0   = blockIdx.x * 16;
  const int i0   = blockIdx.y * NI;

  __shared__ float vbuf[16 * 512];    // v slice for this b-tile: [b_local][o*16+d]
  for (int n = tid; n < 16 * 512; n += 256)
    vbuf[n] = vprev[(size_t)(b0 + (n >> 9)) * 512 + (n & 511)];
  __syncthreads();

  const int o_base = wave * 4;
  for (int ii = 0; ii < NI; ++ii) {
    const int i = i0 + ii;
    const float* xp = x + ((size_t)(b0 + l15) * IN_C + i) * IN_D;
    const v2f a_lo = *(const v2f*)(xp + 2 * h);
    const v2f a_hi = *(const v2f*)(xp + 4 + 2 * h);
#pragma unroll
    for (int j = 0; j < 4; ++j) {
      const int o = o_base + j;
      const float* wp = Wm + (((size_t)i * OUT_C + o) * OUT_D + l15) * IN_D;
      const v2f b_lo = *(const v2f*)(wp + 2 * h);
      const v2f b_hi = *(const v2f*)(wp + 4 + 2 * h);
      v8f tile = {};
      tile = wmma_f32_16x16x4(a_lo, b_lo, tile);
      tile = wmma_f32_16x16x4(a_hi, b_hi, tile);
#pragma unroll
      for (int v = 0; v < 8; ++v) {
        float p = tile[v] * vbuf[(v + 8 * h) * 512 + o * 16 + l15];
        p += __shfl_xor(p, 1, 16);    // reduce over d within each 16-lane half
        p += __shfl_xor(p, 2, 16);
        p += __shfl_xor(p, 4, 16);
        p += __shfl_xor(p, 8, 16);
        if (l15 == v) {               // unique (b,i,o) writer across the grid
          const size_t idx = ((size_t)(b0 + v + 8 * h) * IN_C + i) * OUT_C + o;
          blog[idx] += p;
        }
      }
    }
  }
}

// ---- v = squash(s) --------------------------------------------------------
__global__ void caps_squash(const float* __restrict__ s,
                            float* __restrict__ vout) {
  const int bo = blockIdx.x * blockDim.x + threadIdx.x;  // exactly 2048
  const float* sp = s + (size_t)bo * OUT_D;
  float sv[OUT_D];
  float sq = 0.f;
#pragma unroll
  for (int d = 0; d < OUT_D; ++d) {
    sv[d] = sp[d];
    sq += sv[d] * sv[d];
  }
  const float scale = (sq / (1.0f + sq)) * rsqrtf(sq + 1e-9f);
  float* op = vout + (size_t)bo * OUT_D;
#pragma unroll
  for (int d = 0; d < OUT_D; ++d) op[d] = sv[d] * scale;
}

extern "C" void kernel_launch(void* const* d_in, const int* in_sizes, int n_in,
                              void* d_out, int out_size, void* d_ws,
                              size_t ws_size, hipStream_t stream) {
  const float* x  = (const float*)d_in[0];   // [64,2048,8]
  const float* Wm = (const float*)d_in[1];   // [1,2048,32,16,8]
  float* out = (float*)d_out;                // [64,32,16]

  float* ws   = (float*)d_ws;                // ~16.3 MB total
  float* blog = ws;                                          // 64*2048*32
  float* s    = blog + (size_t)BATCH * IN_C * OUT_C;         // 64*32*16
  float* vbuf = s + (size_t)BATCH * OUT_C * OUT_D;           // 64*32*16
  float* mx   = vbuf + (size_t)BATCH * OUT_C * OUT_D;        // 64*32
  float* invz = mx + BATCH * OUT_C;                          // 64*32

  hipMemsetAsync(blog, 0, (size_t)BATCH * IN_C * OUT_C * sizeof(float), stream);

  for (int t = 0; t < 3; ++t) {
    caps_norms<<<BATCH * OUT_C, 128, 0, stream>>>(blog, mx, invz);
    hipMemsetAsync(s, 0, (size_t)BATCH * OUT_C * OUT_D * sizeof(float), stream);
    caps_s_wmma<<<dim3(BATCH / 16, IN_C / NI), 256, 0, stream>>>(
        x, Wm, blog, mx, invz, s);
    caps_squash<<<(BATCH * OUT_C) / 256, 256, 0, stream>>>(
        s, (t == 2) ? out : vbuf);
    if (t < 2)
      caps_b_wmma<<<dim3(BATCH / 16, IN_C / NI), 256, 0, stream>>>(
          x, Wm, vbuf, blog);
  }
}